// ReinforceDecimator_67628555042888
// MI455X (gfx1250) — compile-verified
//
#include <hip/hip_runtime.h>

// ReinforceDecimator for MI455X (gfx1250, wave32).
// Memory-bound sparse segment workload: streaming edge passes with
// non-temporal hints + global_prefetch, L2-resident V-sized atomic scratch,
// LDS pre-reduction for the B=64 hotspot, and the var_agg @ w_score matvec
// mapped onto v_wmma_f32_16x16x4_f32 (16 variables per wave).

#define ALPHA_F   30.0f
#define EPS_F     1e-10f
#define CONV_TOL_F 0.01f

typedef __attribute__((ext_vector_type(2))) float v2f;
typedef __attribute__((ext_vector_type(8))) float v8f;

__device__ __forceinline__ float sign_f(float s) {
  return (s > 0.0f) ? 1.0f : ((s < 0.0f) ? -1.0f : 0.0f);
}

// ---------------------------------------------------------------------------
// Edge pass 1: x = |prev - fs0| * emask ; segment max into m[V] (x >= 0, so
// int-bit atomicMax is exact) ; segment sum of fs into agg[V,2].
// ---------------------------------------------------------------------------
__global__ __launch_bounds__(256) void edge_pass1(
    const float* __restrict__ fs, const float* __restrict__ prev,
    const float* __restrict__ emask, const int* __restrict__ evid,
    float* __restrict__ m, float* __restrict__ agg, long long E)
{
  const long long stride = (long long)gridDim.x * blockDim.x;
  for (long long i = (long long)blockIdx.x * blockDim.x + threadIdx.x;
       i < E; i += stride) {
    const long long nxt = i + stride;
    if (nxt < E) {                               // gfx1250 global_prefetch_b8
      __builtin_prefetch(&prev[nxt], 0, 0);
      __builtin_prefetch(&fs[2 * nxt], 0, 0);
      __builtin_prefetch(&emask[nxt], 0, 0);
      __builtin_prefetch(&evid[nxt], 0, 0);
    }
    const float f0 = __builtin_nontemporal_load(&fs[2 * i]);
    const float f1 = __builtin_nontemporal_load(&fs[2 * i + 1]);
    const float pv = __builtin_nontemporal_load(&prev[i]);
    const float em = __builtin_nontemporal_load(&emask[i]);
    const int   v  = __builtin_nontemporal_load(&evid[i]);
    const float x  = fabsf(pv - f0) * em;
    atomicMax((int*)(m + v), __float_as_int(x));   // nonneg float max
    atomicAdd(agg + 2 * (long long)v,     f0);
    atomicAdd(agg + 2 * (long long)v + 1, f1);
  }
}

// ---------------------------------------------------------------------------
// Edge pass 2: c = exp(ALPHA*(x - m[v])) ; num += x*c ; den += c.
// m[] is L2-resident (4 MB), edges stream NT.
// ---------------------------------------------------------------------------
__global__ __launch_bounds__(256) void edge_pass2(
    const float* __restrict__ fs, const float* __restrict__ prev,
    const float* __restrict__ emask, const int* __restrict__ evid,
    const float* __restrict__ m, float* __restrict__ num,
    float* __restrict__ den, long long E)
{
  const long long stride = (long long)gridDim.x * blockDim.x;
  for (long long i = (long long)blockIdx.x * blockDim.x + threadIdx.x;
       i < E; i += stride) {
    const long long nxt = i + stride;
    if (nxt < E) {
      __builtin_prefetch(&prev[nxt], 0, 0);
      __builtin_prefetch(&fs[2 * nxt], 0, 0);
      __builtin_prefetch(&emask[nxt], 0, 0);
      __builtin_prefetch(&evid[nxt], 0, 0);
    }
    const float f0 = __builtin_nontemporal_load(&fs[2 * i]);
    const float pv = __builtin_nontemporal_load(&prev[i]);
    const float em = __builtin_nontemporal_load(&emask[i]);
    const int   v  = __builtin_nontemporal_load(&evid[i]);
    const float x  = fabsf(pv - f0) * em;
    const float c  = expf(ALPHA_F * (x - m[v]));
    atomicAdd(num + v, x * c);
    atomicAdd(den + v, c);
  }
}

// ---------------------------------------------------------------------------
// Variable pass: sum_diff = num/(den+eps)*active ; per-batch segment max with
// an LDS pre-reduction (batch ids are sorted -> each block touches ~1-2 of
// the 64 batch slots; avoids 1M global atomics onto 64 addresses).
// ---------------------------------------------------------------------------
__global__ __launch_bounds__(256) void var_pass(
    const float* __restrict__ num, const float* __restrict__ den,
    const float* __restrict__ act, const int* __restrict__ bvid,
    float* __restrict__ pb, int V, int B)
{
  extern __shared__ int sm[];                    // B float-bit slots
  for (int j = threadIdx.x; j < B; j += blockDim.x) sm[j] = 0;
  __syncthreads();
  const int v = blockIdx.x * blockDim.x + threadIdx.x;
  if (v < V) {
    const float sd = num[v] / (den[v] + EPS_F) * act[v];   // >= 0 here
    atomicMax(&sm[bvid[v]], __float_as_int(sd));           // ds_max on LDS
  }
  __syncthreads();
  for (int j = threadIdx.x; j < B; j += blockDim.x) {
    const int bits = sm[j];
    if (bits != 0) atomicMax((int*)(pb + j), bits);
  }
}

// ---------------------------------------------------------------------------
// Score pass: score = agg @ w + b, signs[v] = sign(score), via
// v_wmma_f32_16x16x4_f32. One wave32 -> 16 variables.
//   A(16x4) row m = [agg0(m), agg1(m), 1, 0]   (lanes 0-15: K=0,1; 16-31: K=2,3)
//   B(4x16) col n = [w0, w1, bias, 0]          (VGPR0: K=0/K=2, VGPR1: K=1/K=3)
//   D columns are all equal => lane r (r<8) holds score_r in C[r],
//   lane 16+r holds score_{8+r} in C[r]  -> static register indexing only.
// The tile guard is wave-uniform so EXEC is all-1s at the WMMA.
// ---------------------------------------------------------------------------
__global__ __launch_bounds__(256) void score_sign_wmma(
    const float* __restrict__ agg, const float* __restrict__ w,
    const float* __restrict__ bsc, float* __restrict__ signs, int V)
{
  const int lane = threadIdx.x & 31;
  const long long wave =
      (long long)blockIdx.x * (blockDim.x >> 5) + (threadIdx.x >> 5);
  const long long base = wave * 16;
  if (base >= (long long)V) return;              // wave-uniform
  const float w0 = w[0], w1 = w[1], bias = bsc[0];

  if (base + 16 <= (long long)V) {               // full tile, EXEC all ones
    float a0 = 1.0f, a1 = 0.0f;                  // K=2 -> 1 (bias row), K=3 -> 0
    if (lane < 16) {
      const float* p = agg + 2 * (base + lane);
      a0 = p[0];                                 // K=0
      a1 = p[1];                                 // K=1
    }
    v2f A;  A.x = a0;  A.y = a1;
    v2f Bm; Bm.x = (lane < 16) ? w0 : bias;
            Bm.y = (lane < 16) ? w1 : 0.0f;
    v8f C = {};
    C = __builtin_amdgcn_wmma_f32_16x16x4_f32(
        /*neg_a=*/false, A, /*neg_b=*/false, Bm,
        /*c_mod=*/(short)0, C, /*reuse_a=*/false, /*reuse_b=*/false);
#pragma unroll
    for (int r = 0; r < 8; ++r) {
      const float sg = sign_f(C[r]);
      if (lane == r)      signs[base + r]     = sg;
      if (lane == 16 + r) signs[base + 8 + r] = sg;
    }
  } else {                                        // scalar tail (V%16 != 0)
    if (lane < 16 && base + lane < (long long)V) {
      const float* p = agg + 2 * (base + lane);
      const float s = fmaf(p[0], w0, fmaf(p[1], w1, bias));
      signs[base + lane] = sign_f(s);
    }
  }
}

// ---------------------------------------------------------------------------
// Batch pass: active_mask update (B=64, trivial).
// ---------------------------------------------------------------------------
__global__ void batch_pass(const float* __restrict__ pb,
                           const float* __restrict__ am_in,
                           float* __restrict__ am_out, int B)
{
  const int b = blockIdx.x * blockDim.x + threadIdx.x;
  if (b < B) am_out[b] = (pb[b] <= CONV_TOL_F) ? 0.0f : am_in[b];
}

// ---------------------------------------------------------------------------
// Edge pass 3: fs_out[:,0] = fs[:,0];
//              fs_out[:,1] = mask_e * sign + (1-mask_e) * fs[:,1].
// ---------------------------------------------------------------------------
__global__ __launch_bounds__(256) void edge_pass3(
    const float* __restrict__ fs, const int* __restrict__ evid,
    const int* __restrict__ bvid, const float* __restrict__ signs,
    const float* __restrict__ am, float* __restrict__ out_fs, long long E)
{
  const long long stride = (long long)gridDim.x * blockDim.x;
  for (long long i = (long long)blockIdx.x * blockDim.x + threadIdx.x;
       i < E; i += stride) {
    const long long nxt = i + stride;
    if (nxt < E) {
      __builtin_prefetch(&fs[2 * nxt], 0, 0);
      __builtin_prefetch(&evid[nxt], 0, 0);
    }
    const float f0 = __builtin_nontemporal_load(&fs[2 * i]);
    const float f1 = __builtin_nontemporal_load(&fs[2 * i + 1]);
    const int   v  = __builtin_nontemporal_load(&evid[i]);
    const float mk = am[bvid[v]];                 // L2-resident gathers
    const float sg = signs[v];
    __builtin_nontemporal_store(f0, &out_fs[2 * i]);
    __builtin_nontemporal_store(mk * sg + (1.0f - mk) * f1,
                                &out_fs[2 * i + 1]);
  }
}

// ---------------------------------------------------------------------------
extern "C" void kernel_launch(void* const* d_in, const int* in_sizes, int n_in,
                              void* d_out, int out_size, void* d_ws,
                              size_t ws_size, hipStream_t stream)
{
  const float* var_state = (const float*)d_in[0];   // (E,3)
  const float* fs        = (const float*)d_in[1];   // (E,2)
  const float* prev      = (const float*)d_in[2];   // (E,)
  const float* am_in     = (const float*)d_in[3];   // (B,1)
  const float* act_var   = (const float*)d_in[4];   // (V,1)
  const float* emask     = (const float*)d_in[5];   // (E,1)
  const float* w         = (const float*)d_in[6];   // (2,1)
  const float* bsc       = (const float*)d_in[7];   // (1,)
  const int*   evid      = (const int*)d_in[8];     // (E,)
  const int*   bvid      = (const int*)d_in[9];     // (V,)

  const long long E = in_sizes[2];
  const int       V = in_sizes[4];
  const int       B = in_sizes[3];

  // Scratch layout: [ m(V) | num(V) | den(V) | agg(2V) | signs(V) | pb(B) ]
  float* ws    = (float*)d_ws;
  float* m     = ws;
  float* num   = ws + (size_t)V;
  float* den   = ws + 2 * (size_t)V;
  float* agg   = ws + 3 * (size_t)V;
  float* signs = ws + 5 * (size_t)V;
  float* pb    = ws + 6 * (size_t)V;

  // Output layout: [ variable_state(3E) | function_state(2E) | active_mask(B) ]
  float* out_vs = (float*)d_out;
  float* out_fs = out_vs + 3 * (size_t)E;
  float* out_am = out_fs + 2 * (size_t)E;

  // Zero the atomic scratch (max identities are 0.0f bits; values are >= 0).
  hipMemsetAsync(d_ws, 0, ((size_t)6 * V + B) * sizeof(float), stream);
  // variable_state passes through unchanged.
  hipMemcpyAsync(out_vs, var_state, (size_t)3 * E * sizeof(float),
                 hipMemcpyDeviceToDevice, stream);

  const int TB = 256;                 // 8 wave32s per block
  const int eBlocks = 8192;           // grid-stride over 8M edges

  edge_pass1<<<eBlocks, TB, 0, stream>>>(fs, prev, emask, evid, m, agg, E);
  edge_pass2<<<eBlocks, TB, 0, stream>>>(fs, prev, emask, evid, m, num, den, E);

  const int vBlocks = (V + TB - 1) / TB;
  var_pass<<<vBlocks, TB, (size_t)B * sizeof(int), stream>>>(
      num, den, act_var, bvid, pb, V, B);

  const long long waves  = ((long long)V + 15) / 16;
  const int       sBlocks = (int)((waves + 7) / 8);
  score_sign_wmma<<<sBlocks, TB, 0, stream>>>(agg, w, bsc, signs, V);

  batch_pass<<<(B + 63) / 64, 64, 0, stream>>>(pb, am_in, out_am, B);

  edge_pass3<<<eBlocks, TB, 0, stream>>>(fs, evid, bvid, signs, out_am,
                                         out_fs, E);
}